// HypergraphAttentionLayer_57801669869838
// MI455X (gfx1250) — compile-verified
//
#include <hip/hip_runtime.h>

typedef __attribute__((ext_vector_type(2))) float v2f;
typedef __attribute__((ext_vector_type(8))) float v8f;

#define F_IN   128
#define F_OUT  64
#define DEG    32
#define WAVES_PER_BLOCK 8
#define LDS_STRIDE 132   // 128 + 4 DWORDs pad -> conflict-free ds_load_b64

// Kernel 1: h = x @ W^T via V_WMMA_F32_16X16X4_F32, fused s1 = h@a1, s2 = h@a2.
// W (32KB) is staged once per workgroup into LDS; each wave computes a
// 16-row x 64-col tile of h (4 accumulators of 16x16).
__global__ __launch_bounds__(256) void hga_gemm_scores(
    const float* __restrict__ x,      // (N, 128)
    const float* __restrict__ W,      // (64, 128) row-major; B[k][n] = W[n][k]
    const float* __restrict__ a,      // (128,)  a1 = a[0:64], a2 = a[64:128]
    float* __restrict__ h,            // (N, 64)
    float* __restrict__ s1,           // (N,)
    float* __restrict__ s2,           // (N,)
    int Nrows)
{
    __shared__ float Wl[F_OUT * LDS_STRIDE];   // 33 KB of the 320 KB WGP LDS

    // ---- cooperative staging of W (row-major, padded stride), float4 chunks
    for (int c = threadIdx.x; c < F_OUT * (F_IN / 4); c += 256) {
        const int row = c >> 5;            // c / 32
        const int kc  = (c & 31) << 2;     // (c % 32) * 4
        const float4 w4 = ((const float4*)(W + (size_t)row * F_IN))[kc >> 2];
        float* dst = &Wl[row * LDS_STRIDE + kc];
        dst[0] = w4.x; dst[1] = w4.y; dst[2] = w4.z; dst[3] = w4.w;
    }
    __syncthreads();

    const int lane = threadIdx.x & 31;
    const int wave = threadIdx.x >> 5;
    const int tile = blockIdx.x * WAVES_PER_BLOCK + wave;   // 16-row tile id
    const int row_base = tile * 16;
    if (row_base >= Nrows) return;                          // wave-uniform exit

    const int half = lane >> 4;       // 0: lanes 0-15, 1: lanes 16-31
    const int l16  = lane & 15;
    const int koff = half * 2;        // K offset within 4-wide K slab

    // A: this lane supplies row (row_base + l16), elements K = kb+koff, kb+koff+1
    const float* __restrict__ xrow = x + (size_t)(row_base + l16) * F_IN;

    v8f acc0 = {0,0,0,0,0,0,0,0};
    v8f acc1 = {0,0,0,0,0,0,0,0};
    v8f acc2 = {0,0,0,0,0,0,0,0};
    v8f acc3 = {0,0,0,0,0,0,0,0};

    // B column supplied by this lane for each of the 4 col-tiles: n = t*16 + l16
    const float* w0 = &Wl[(0 * 16 + l16) * LDS_STRIDE];
    const float* w1 = &Wl[(1 * 16 + l16) * LDS_STRIDE];
    const float* w2 = &Wl[(2 * 16 + l16) * LDS_STRIDE];
    const float* w3 = &Wl[(3 * 16 + l16) * LDS_STRIDE];

    #pragma unroll 4
    for (int kb = 0; kb < F_IN; kb += 4) {
        v2f av, b0, b1, b2, b3;
        av.x = xrow[kb + koff];  av.y = xrow[kb + koff + 1];
        b0.x = w0[kb + koff];    b0.y = w0[kb + koff + 1];
        b1.x = w1[kb + koff];    b1.y = w1[kb + koff + 1];
        b2.x = w2[kb + koff];    b2.y = w2[kb + koff + 1];
        b3.x = w3[kb + koff];    b3.y = w3[kb + koff + 1];
        acc0 = __builtin_amdgcn_wmma_f32_16x16x4_f32(false, av, false, b0, (short)0, acc0, false, false);
        acc1 = __builtin_amdgcn_wmma_f32_16x16x4_f32(false, av, false, b1, (short)0, acc1, false, false);
        acc2 = __builtin_amdgcn_wmma_f32_16x16x4_f32(false, av, false, b2, (short)0, acc2, false, false);
        acc3 = __builtin_amdgcn_wmma_f32_16x16x4_f32(false, av, false, b3, (short)0, acc3, false, false);
    }

    // C/D layout: VGPR v holds h[M][N]: lanes 0-15 -> M=v,   N=l16
    //                                   lanes 16-31 -> M=v+8, N=l16
    const int mrow = row_base + half * 8;

    // Store h tile
    #pragma unroll
    for (int v = 0; v < 8; ++v) {
        float* hrow = h + (size_t)(mrow + v) * F_OUT + l16;
        hrow[0]  = acc0[v];
        hrow[16] = acc1[v];
        hrow[32] = acc2[v];
        hrow[48] = acc3[v];
    }

    // Fused scores: per-lane partial dot over its column, then 16-lane xor reduction
    float p1[8], p2[8];
    const float a10 = a[0 * 16 + l16],      a11 = a[1 * 16 + l16];
    const float a12 = a[2 * 16 + l16],      a13 = a[3 * 16 + l16];
    const float a20 = a[64 + 0 * 16 + l16], a21 = a[64 + 1 * 16 + l16];
    const float a22 = a[64 + 2 * 16 + l16], a23 = a[64 + 3 * 16 + l16];
    #pragma unroll
    for (int v = 0; v < 8; ++v) {
        p1[v] = acc0[v] * a10 + acc1[v] * a11 + acc2[v] * a12 + acc3[v] * a13;
        p2[v] = acc0[v] * a20 + acc1[v] * a21 + acc2[v] * a22 + acc3[v] * a23;
    }
    #pragma unroll
    for (int m = 8; m >= 1; m >>= 1) {
        #pragma unroll
        for (int v = 0; v < 8; ++v) {
            p1[v] += __shfl_xor(p1[v], m, 32);
            p2[v] += __shfl_xor(p2[v], m, 32);
        }
    }
    if (l16 == 0) {
        #pragma unroll
        for (int v = 0; v < 8; ++v) {
            s1[mrow + v] = p1[v];
            s2[mrow + v] = p2[v];
        }
    }
}

// Kernel 2: one wave per hyperedge (DEG=32 incidences, lane = rank within edge).
// pair-mean attention:  sum_r  s1[n_r]*(DEG-1-r) + s2[n_r]*r,  / (DEG*(DEG-1)/2)
__global__ __launch_bounds__(256) void hga_edge_attn(
    const int* __restrict__ nodes,    // hyperedge_index[0], (E,)
    const float* __restrict__ s1,
    const float* __restrict__ s2,
    float* __restrict__ eatt,         // (H,)
    int Hn)
{
    const int lane = threadIdx.x & 31;
    const int e = blockIdx.x * WAVES_PER_BLOCK + (threadIdx.x >> 5);
    if (e >= Hn) return;

    const int node = nodes[(size_t)e * DEG + lane];
    float v = s1[node] * (float)(DEG - 1 - lane) + s2[node] * (float)lane;
    #pragma unroll
    for (int m = 16; m >= 1; m >>= 1) v += __shfl_xor(v, m, 32);

    if (lane == 0) {
        const float npairs = (float)((DEG * (DEG - 1)) / 2);   // 496
        eatt[e] = v / npairs;
    }
}

extern "C" void kernel_launch(void* const* d_in, const int* in_sizes, int n_in,
                              void* d_out, int out_size, void* d_ws, size_t ws_size,
                              hipStream_t stream)
{
    const float* x = (const float*)d_in[0];    // (N,128)
    const float* W = (const float*)d_in[1];    // (64,128)
    const float* a = (const float*)d_in[2];    // (128,)
    const int* hidx = (const int*)d_in[3];     // (2,E) -> row0 = nodes

    const int N = in_sizes[0] / F_IN;          // 200000
    const int E = in_sizes[3] / 2;             // 1,600,000
    const int Hn = E / DEG;                    // 50000

    float* h    = (float*)d_out;               // (N,64)
    float* eatt = (float*)d_out + (size_t)N * F_OUT;  // (H,)
    float* s1   = (float*)d_ws;                // (N,)
    float* s2   = s1 + N;                      // (N,)

    const int tiles = (N + 15) / 16;                                      // 12500
    const int blocks1 = (tiles + WAVES_PER_BLOCK - 1) / WAVES_PER_BLOCK;  // 1563
    hga_gemm_scores<<<blocks1, 256, 0, stream>>>(x, W, a, h, s1, s2, N);

    const int blocks2 = (Hn + WAVES_PER_BLOCK - 1) / WAVES_PER_BLOCK;     // 6250
    hga_edge_attn<<<blocks2, 256, 0, stream>>>(hidx, s1, s2, eatt, Hn);
}